// Calibration_78606491451591
// MI455X (gfx1250) — compile-verified
//
#include <hip/hip_runtime.h>

// ---------------------------------------------------------------------------
// Calibration kernel for MI455X (gfx1250, wave32).
//
// Heavy part = argmin over a (8192 x 1024) squared-distance matrix per batch.
// Entire d2 = |o|^2 + |bd|^2 - 2 o.bd packed into one K=4 f32 WMMA:
//   A[M][*] = [-2*ox, -2*oy, 1, ox^2+oy^2]   (16x4, per-wave, 16 points)
//   B[*][N] = [bdx, bdy, bdx^2+bdy^2, 1]     (4x16, chunk of bounds)
//   D = A x B  ==  d2 tile (16x16, f32)      via V_WMMA_F32_16X16X4_F32
//
// Round-2 changes vs round-1:
//  * B fragments pre-swizzled into per-lane WMMA layout in LDS -> the inner
//    loop is one unconditional ds_load_b64 + wmma + cndmask chain (no EXEC
//    branching; round-1 codegen had s_and_saveexec per iteration).
//  * chunk loop unrolled x4 so 4 independent wmma/ds chains pipeline.
//  * cross-lane argmin via branch-free u64 min on (monotone-float-key, idx)
//    instead of the branchy (val,idx) shfl combine.
//
// ISA fragment layouts (CDNA5 ISA §7.12.2, wave32):
//   A 16x4 f32 : lanes 0-15 -> {K0,K1} for row M=lane ; lanes 16-31 -> {K2,K3}
//   B 4x16 f32 : lanes 0-15 -> (B[0][n], B[1][n]) ; lanes 16-31 -> (B[2][n], B[3][n])
//   C/D 16x16  : vgpr r, lanes 0-15 -> M=r, N=lane ; lanes 16-31 -> M=r+8, N=lane-16
// ---------------------------------------------------------------------------

typedef __attribute__((ext_vector_type(2))) float v2f;
typedef __attribute__((ext_vector_type(8))) float v8f;

#define IMG    224
#define NB     16          // batch
#define NV     8           // views
#define NPTS   8192        // points per batch
#define NM     1024        // bounds per view
#define NCHUNK (NM / 16)   // 64 WMMA column-chunks

__global__ __launch_bounds__(256) void calib_wmma_kernel(
    const float* __restrict__ pc,         // (16, 8192, 3)
    const float* __restrict__ mask,       // (16, 8, 224, 224)
    const float* __restrict__ bounds,     // (16, 8, 1024, 2)
    const float* __restrict__ inv_param,  // (16, 8, 4, 4)
    const float* __restrict__ proj_fine,  // (16, 8, 8192, 2)
    const float* __restrict__ proj_finez, // (16, 8, 8192)
    const int*   __restrict__ view_id_p,  // scalar
    float*       __restrict__ out)        // (16, 8192, 3)
{
    // B fragments, pre-swizzled to per-lane WMMA layout: s_bf[chunk*32 + lane]
    __shared__ v2f s_bf[NCHUNK * 32];     // 16 KB
    __shared__ int s_nn[128];             // 8 waves * 16 points

    const int v    = *view_id_p;
    const int tid  = threadIdx.x;
    const int lane = tid & 31;
    const int wave = tid >> 5;
    const int hlan = lane & 15;           // position within 16-lane group
    const bool hi  = lane >= 16;

    const int blocksPerBatch = NPTS / 128;               // 64
    const int b        = blockIdx.x / blocksPerBatch;
    const int tileBase = (blockIdx.x % blocksPerBatch) * 128;

    const float invImg = 1.0f / (float)IMG;

    // ---- stage B fragments of (b, v) into LDS, WMMA-lane layout ---------
    const float* bptr = bounds + ((size_t)(b * NV + v) * NM) * 2;
    for (int e = tid; e < NCHUNK * 32; e += 256) {
        const int l   = e & 31;
        const int col = (e >> 5) * 16 + (l & 15);
        const float bx = bptr[col * 2 + 0] * invImg;
        const float by = bptr[col * 2 + 1] * invImg;
        v2f f;
        if (l < 16) { f.x = bx;                 f.y = by;   }  // rows K0,K1
        else        { f.x = bx * bx + by * by;  f.y = 1.0f; }  // rows K2,K3
        s_bf[e] = f;
    }
    __syncthreads();

    // ---- per-wave A fragment: 16 points ---------------------------------
    const int p = tileBase + wave * 16 + hlan;           // this lane's point row
    const float* pf = proj_fine + ((size_t)(b * NV + v) * NPTS + p) * 2;
    const float px = pf[0];
    const float py = pf[1];
    const float fx = rintf(px);                          // pxy_i[...,0]
    const float fy = rintf((float)IMG - py);             // pxy_i[...,1]
    const float ox = fx * invImg;
    const float oy = fy * invImg;
    const float o2 = ox * ox + oy * oy;

    v2f afrag;
    afrag.x = hi ? 1.0f : (-2.0f * ox);                  // K2 : K0
    afrag.y = hi ? o2   : (-2.0f * oy);                  // K3 : K1

    // ---- sweep bounds: 64 chunks, branch-free inner loop ----------------
    float bestd[8];
    int   besti[8];
#pragma unroll
    for (int r = 0; r < 8; ++r) { bestd[r] = 3.0e38f; besti[r] = 0; }

#pragma unroll 4
    for (int c = 0; c < NCHUNK; ++c) {
        const v2f bfrag = s_bf[c * 32 + lane];           // one ds_load_b64
        const int col   = c * 16 + hlan;                 // this lane's C column

        v8f acc = {};
        // 8 args: (neg_a, A, neg_b, B, c_mod, C, reuse_a, reuse_b)
        acc = __builtin_amdgcn_wmma_f32_16x16x4_f32(
                false, afrag, false, bfrag, (short)0, acc, false, false);

#pragma unroll
        for (int r = 0; r < 8; ++r) {                    // row M = r (+8 for hi lanes)
            const float d = acc[r];
            if (d < bestd[r]) { bestd[r] = d; besti[r] = col; }  // strict <: first hit wins
        }
    }

    // ---- cross-lane argmin: branch-free u64 min over (key, idx) ---------
    // monotone float->uint map (valid for negatives from wmma rounding too)
#pragma unroll
    for (int r = 0; r < 8; ++r) {
        unsigned int db = __float_as_uint(bestd[r]);
        db ^= (unsigned int)(((int)db >> 31)) | 0x80000000u;
        unsigned long long key =
            ((unsigned long long)db << 32) | (unsigned int)besti[r];
#pragma unroll
        for (int mstep = 1; mstep < 16; mstep <<= 1) {   // masks keep 16-lane groups
            const unsigned long long ok = __shfl_xor(key, mstep, 32);
            key = (ok < key) ? ok : key;                 // ties -> smaller idx
        }
        besti[r] = (int)(unsigned int)key;
    }

    // lane 0 owns rows 0..7, lane 16 owns rows 8..15 -> stash in LDS
    if (lane == 0) {
#pragma unroll
        for (int r = 0; r < 8; ++r) s_nn[wave * 16 + r] = besti[r];
    }
    if (lane == 16) {
#pragma unroll
        for (int r = 0; r < 8; ++r) s_nn[wave * 16 + 8 + r] = besti[r];
    }
    __syncthreads();

    // ---- finish: lanes 0..15 each complete their own point --------------
    if (!hi) {
        const int nn = s_nn[wave * 16 + hlan];

        const float* bnd = bounds + ((size_t)(b * NV + v) * NM + nn) * 2;
        const float nbx = bnd[0];
        const float nby = bnd[1];
        const float z   = proj_finez[(size_t)(b * NV + v) * NPTS + p];

        // back = [nbx*z, nby*z, z, 1] @ inv_param[b, v]  (first 3 cols)
        const float* ip = inv_param + (size_t)(b * NV + v) * 16;
        const float h0 = nbx * z, h1 = nby * z;
        const float back0 = h0 * ip[0] + h1 * ip[4] + z * ip[8]  + ip[12];
        const float back1 = h0 * ip[1] + h1 * ip[5] + z * ip[9]  + ip[13];
        const float back2 = h0 * ip[2] + h1 * ip[6] + z * ip[10] + ip[14];

        // padded-mask lookup: mv = pad(mask, 1) ; zero border => bounds test
        const int ix = (int)fx;
        const int iy = (int)fy;
        const int xi = min(max(iy + 1, 0), IMG + 1);
        const int yi = min(max(ix + 1, 0), IMG + 1);
        float mval = 0.0f;
        if (xi >= 1 && xi <= IMG && yi >= 1 && yi <= IMG)
            mval = mask[((size_t)(b * NV + v) * IMG + (xi - 1)) * IMG + (yi - 1)];
        const bool outm = (mval == 0.0f);

        const float* pcp = pc  + ((size_t)b * NPTS + p) * 3;
        float*       op  = out + ((size_t)b * NPTS + p) * 3;
        op[0] = outm ? back0 : pcp[0];
        op[1] = outm ? back1 : pcp[1];
        op[2] = outm ? back2 : pcp[2];
    }
}

extern "C" void kernel_launch(void* const* d_in, const int* in_sizes, int n_in,
                              void* d_out, int out_size, void* d_ws, size_t ws_size,
                              hipStream_t stream) {
    const float* pc         = (const float*)d_in[0];
    const float* mask       = (const float*)d_in[1];
    const float* bounds     = (const float*)d_in[2];
    const float* inv_param  = (const float*)d_in[3];
    const float* proj_fine  = (const float*)d_in[4];
    const float* proj_finez = (const float*)d_in[5];
    const int*   view_id    = (const int*)d_in[6];

    dim3 grid(NB * (NPTS / 128));   // 1024 blocks: 64 blocks per batch
    dim3 block(256);                // 8 waves x 16 points = 128 points per block
    hipLaunchKernelGGL(calib_wmma_kernel, grid, block, 0, stream,
                       pc, mask, bounds, inv_param, proj_fine, proj_finez,
                       view_id, (float*)d_out);
}